// OCTMambaBlock_71021579207190
// MI455X (gfx1250) — compile-verified
//
#include <hip/hip_runtime.h>
#include <math.h>

// ---------------------------------------------------------------------------
// OCTMambaBlock for MI455X (gfx1250, wave32).
//  * All GEMM-like stages -> V_WMMA_F32_16X16X4_F32 (fp32 in/acc: the block is
//    ~9 GFLOP with a <150MB working set, L2-resident at 192MB, so it is
//    latency/BW bound, not matrix-throughput bound -> fp32 WMMA is free).
//  * GEMM weight panels staged to LDS via the Tensor Data Mover
//    (tensor_load_to_lds + s_wait_tensorcnt), B-fragments read via DS.
//  * All conditional loads are branch-free (clamped address * 0/1 mask) to
//    avoid per-load s_and_saveexec serialization seen in round 1.
// ---------------------------------------------------------------------------

typedef float v2f __attribute__((ext_vector_type(2)));
typedef float v8f __attribute__((ext_vector_type(8)));
typedef unsigned int uv4 __attribute__((ext_vector_type(4)));
typedef int iv4 __attribute__((ext_vector_type(4)));
typedef int iv8 __attribute__((ext_vector_type(8)));

#define NB    2
#define C_IN  32
#define C_OUT 64
#define SW    32
#define SH    32
#define SD    16
#define LSP   (SW*SH*SD)      // 16384 spatial positions per batch
#define NROW  (NB*LSP)        // 32768 token rows
#define DIN   128
#define DST   16
#define DTR   4
#define EPSV  1e-5f

#if __has_builtin(__builtin_amdgcn_tensor_load_to_lds)
  #define TDM_OK 1
  #if __has_include(<hip/amd_detail/amd_gfx1250_TDM.h>)
    #define TDM_ARGS6 1     // amdgpu-toolchain (clang-23): 6-arg builtin
  #endif
#endif

__device__ __forceinline__ v8f wmma4(v2f a, v2f b, v8f c) {
  // (neg_a, A, neg_b, B, c_mod, C, reuse_a, reuse_b)
  return __builtin_amdgcn_wmma_f32_16x16x4_f32(false, a, false, b, (short)0, c,
                                               false, false);
}

// Stage a dense fp32 panel of `total` elements into LDS (offset 0) using the
// Tensor Data Mover (1-D tile descriptor), then barrier.  Fallback: manual
// cooperative copy.  Call from ALL waves before any early return.
__device__ __forceinline__ void stage_lds(float* wlds, const float* src, int total)
{
#if defined(TDM_OK)
  if (threadIdx.x < 32) {                       // one TDM issue per block
    unsigned long long ga = (unsigned long long)(const void*)src;
    uv4 g0;
    g0.x = 1u;                                  // count=1, user descriptor
    g0.y = 0u;                                  // lds_addr = 0 (wlds is at 0)
    g0.z = (unsigned)(ga & 0xffffffffull);
    g0.w = (unsigned)((ga >> 32) & 0x1ffffffull) | (2u << 30);  // type=2
    iv8 g1;
    unsigned tot = (unsigned)total;
    g1[0] = (int)(2u << 16);                    // data_size=2 (4 bytes)
    g1[1] = (int)((tot & 0xffffu) << 16);       // tensor_dim0 [79:48] lo
    g1[2] = (int)(((tot >> 16) & 0xffffu) | (1u << 16)); // dim0 hi, dim1=1
    g1[3] = (int)((tot & 0xffffu) << 16);       // tile_dim0 [127:112]
    g1[4] = 0;                                  // tile_dim1=0 (1-D tile)
    g1[5] = (int)tot;                           // dim0 stride (benign)
    g1[6] = 0; g1[7] = 0;
    iv4 z4 = {0, 0, 0, 0};
  #if defined(TDM_ARGS6)
    iv8 z8 = {0, 0, 0, 0, 0, 0, 0, 0};
    __builtin_amdgcn_tensor_load_to_lds(g0, g1, z4, z4, z8, 0);
  #else
    __builtin_amdgcn_tensor_load_to_lds(g0, g1, z4, z4, 0);
  #endif
  #if __has_builtin(__builtin_amdgcn_s_wait_tensorcnt)
    __builtin_amdgcn_s_wait_tensorcnt(0);
  #else
    asm volatile("s_wait_tensorcnt 0x0" ::: "memory");
  #endif
  }
#else
  for (int i = threadIdx.x; i < total; i += (int)blockDim.x)
    wlds[i] = src[i];
#endif
  __syncthreads();
}

// ---------------------------------------------------------------------------
// fp32 WMMA GEMM:  C[M,N] = act(bn(A[M,K] @ Wt[N,K]^T + bias)), ldw == K.
// Weight panel (N*K <= 16384 floats) staged to LDS by TDM; B-frags from DS.
// One wave = one 16x16 tile; block = 128 threads (4 waves across N tiles).
// ---------------------------------------------------------------------------
template <bool NGUARD>
__global__ void gemm_f32_wmma(const float* __restrict__ A, int lda,
                              const float* __restrict__ Wt,
                              float* __restrict__ C, int ldc,
                              int N, int K,
                              const float* __restrict__ bias,
                              const float* __restrict__ bn_g,
                              const float* __restrict__ bn_b,
                              const float* __restrict__ bn_m,
                              const float* __restrict__ bn_v,
                              int relu)
{
  __shared__ float wlds[16384];                 // 64 KB weight panel
  stage_lds(wlds, Wt, N * K);

  const int lane = threadIdx.x & 31;
  const int wave = threadIdx.x >> 5;
  const int tm   = blockIdx.x;
  const int tn   = blockIdx.y * 4 + wave;
  if (tn * 16 >= N) return;                     // wave-uniform

  const int mrow = tm * 16 + (lane & 15);
  const int ncol = tn * 16 + (lane & 15);
  const int koff = (lane < 16) ? 0 : 2;         // A/B lanes0-15:K0/1, 16-31:K2/3

  bool  nok = true; int ncs = ncol; float msk = 1.f;
  if constexpr (NGUARD) {
    nok = (ncol < N);
    ncs = nok ? ncol : 0;                       // clamp -> branch-free DS reads
    msk = nok ? 1.f : 0.f;
  }

  const float* arow = A + (long)mrow * lda + koff;
  const float* brow = wlds + (long)ncs * K + koff;
  v8f acc = {};
  for (int k = 0; k < K; k += 4) {
    v2f a, b;
    a.x = arow[k];     a.y = arow[k + 1];       // global_load_b64
    b.x = brow[k] * msk; b.y = brow[k + 1] * msk; // ds_load_b64
    acc = wmma4(a, b, acc);
  }

  float sc = 1.f, sh = 0.f;
  if (bn_g) {
    float s = bn_g[ncs] * rsqrtf(bn_v[ncs] + EPSV);
    sc = s; sh = bn_b[ncs] - bn_m[ncs] * s;
  }
  const float addb = bias ? bias[ncs] : 0.f;

  union { v8f v; float f[8]; } u; u.v = acc;
  const int mshift = (lane >= 16) ? 8 : 0;
  for (int r = 0; r < 8; ++r) {
    if (!NGUARD || nok) {
      int   mm = tm * 16 + r + mshift;          // C/D layout: VGPR r -> M=r(+8)
      float v  = (u.f[r] + addb) * sc + sh;
      if (relu) v = fmaxf(v, 0.f);
      C[(long)mm * ldc + ncol] = v;
    }
  }
}

// --------- prep: x -> XT[row][ci]  and  conv_w -> W2[tap][ci][co] -----------
__global__ void xpose_kernel(const float* __restrict__ X, float* __restrict__ XT)
{
  int idx = blockIdx.x * 256 + threadIdx.x;     // NB*C_IN*LSP
  int sp = idx % LSP;
  int ci = (idx / LSP) % C_IN;
  int b  = idx / (C_IN * LSP);
  XT[((long)b * LSP + sp) * C_IN + ci] = X[idx];
}

__global__ void wpose_kernel(const float* __restrict__ Wc, float* __restrict__ W2)
{
  int idx = blockIdx.x * 256 + threadIdx.x;     // C_OUT*C_IN*27 = 55296
  if (idx >= C_OUT * C_IN * 27) return;
  int t  = idx % 27;
  int ci = (idx / 27) % C_IN;
  int co = idx / (27 * C_IN);
  W2[((long)t * C_IN + ci) * C_OUT + co] = Wc[idx];
}

// ---------------------------------------------------------------------------
// conv3d 3x3x3 (32->64), pad 1, as 27-tap implicit GEMM over XT/W2.
// A loads: one b64 per lane per k-step; halo handled by clamp+mask (no exec
// divergence).  27 taps x 8 k-steps = 216 v_wmma per wave.
// ---------------------------------------------------------------------------
__global__ void conv3d_wmma(const float* __restrict__ XT,
                            const float* __restrict__ W2,
                            const float* __restrict__ bias,
                            float* __restrict__ Out)
{
  const int lane = threadIdx.x & 31;
  const int wave = threadIdx.x >> 5;
  const int tm   = blockIdx.x;
  const int mrow = tm * 16 + (lane & 15);
  const int b    = mrow / LSP;
  const int sp   = mrow % LSP;
  const int w0   = sp / (SH * SD);
  const int h0   = (sp / SD) % SH;
  const int d0   = sp % SD;
  const int co   = wave * 16 + (lane & 15);
  const int koff = (lane < 16) ? 0 : 2;

  v8f acc = {};
  for (int t = 0; t < 27; ++t) {
    const int  tw = t / 9 - 1, th = (t / 3) % 3 - 1, td = t % 3 - 1;
    const int  ww = w0 + tw, hh = h0 + th, dd = d0 + td;
    const bool inb = (unsigned)ww < SW && (unsigned)hh < SH && (unsigned)dd < SD;
    const int  spc = inb ? (ww * SH + hh) * SD + dd : sp;   // clamped (valid)
    const float am = inb ? 1.f : 0.f;
    const float* arow = XT + ((long)b * LSP + spc) * C_IN + koff;
    const float* wrow = W2 + (long)t * C_IN * C_OUT + co;
    for (int k = 0; k < C_IN; k += 4) {
      v2f a, bb;
      a.x  = arow[k] * am;                       // global_load_b64 + mul
      a.y  = arow[k + 1] * am;
      bb.x = wrow[(long)(k + koff) * C_OUT];     // lane-coalesced
      bb.y = wrow[(long)(k + koff + 1) * C_OUT];
      acc = wmma4(a, bb, acc);
    }
  }

  union { v8f v; float f[8]; } u; u.v = acc;
  const float bv = bias[co];
  const int mshift = (lane >= 16) ? 8 : 0;
  for (int r = 0; r < 8; ++r) {
    int mm = tm * 16 + r + mshift;
    int b2 = mm / LSP, sp2 = mm % LSP;
    Out[((long)b2 * C_OUT + co) * LSP + sp2] = u.f[r] + bv;
  }
}

// ----------------------- GroupNorm (32 groups of 2 ch) ----------------------
__global__ void gn_stats(const float* __restrict__ A,
                         float* __restrict__ mean, float* __restrict__ rstd)
{
  __shared__ float s1[256], s2[256];
  const int pid = blockIdx.x;                   // 0..63 = (b, group)
  const int b = pid / 32, g = pid % 32;
  const int cbase = g * 2;
  float acc = 0.f, acc2 = 0.f;
  for (int i = threadIdx.x; i < 2 * LSP; i += 256) {
    int   c  = cbase + (i >= LSP);
    int   sp = i & (LSP - 1);
    float v  = A[((long)b * C_OUT + c) * LSP + sp];
    acc += v; acc2 += v * v;
  }
  s1[threadIdx.x] = acc; s2[threadIdx.x] = acc2;
  __syncthreads();
  for (int o = 128; o; o >>= 1) {
    if (threadIdx.x < o) {
      s1[threadIdx.x] += s1[threadIdx.x + o];
      s2[threadIdx.x] += s2[threadIdx.x + o];
    }
    __syncthreads();
  }
  if (threadIdx.x == 0) {
    float mu  = s1[0] / (float)(2 * LSP);
    float var = s2[0] / (float)(2 * LSP) - mu * mu;
    mean[pid] = mu;
    rstd[pid] = rsqrtf(var + EPSV);
  }
}

__global__ void gn_apply(float* __restrict__ A, const float* __restrict__ mean,
                         const float* __restrict__ rstd,
                         const float* __restrict__ g, const float* __restrict__ bp)
{
  int   idx = blockIdx.x * 256 + threadIdx.x;
  int   c   = (idx / LSP) % C_OUT;
  int   b   = idx / (C_OUT * LSP);
  int   grp = b * 32 + c / 2;
  float v   = A[idx];
  v = (v - mean[grp]) * rstd[grp] * g[c] + bp[c];
  A[idx] = fmaxf(v, 0.f);                       // relu; A becomes 'identity'
}

// --------------- SFE: depthwise 3x3x3 + BN1  +  pointwise + BN2 + x ---------
__global__ void sfe_kernel(const float* __restrict__ A, float* __restrict__ Out,
                           const float* __restrict__ dw,
                           const float* __restrict__ g1, const float* __restrict__ b1,
                           const float* __restrict__ m1, const float* __restrict__ v1,
                           const float* __restrict__ pw,
                           const float* __restrict__ g2, const float* __restrict__ b2,
                           const float* __restrict__ m2, const float* __restrict__ v2)
{
  int idx = blockIdx.x * 256 + threadIdx.x;
  int sp  = idx % LSP;
  int c   = (idx / LSP) % C_OUT;
  int b   = idx / (C_OUT * LSP);
  int w0 = sp / (SH * SD), h0 = (sp / SD) % SH, d0 = sp % SD;
  const float* wc   = dw + c * 27;
  const float* base = A + ((long)b * C_OUT + c) * LSP;
  float acc = 0.f;
#pragma unroll
  for (int t = 0; t < 27; ++t) {
    int tw = t / 9 - 1, th = (t / 3) % 3 - 1, td = t % 3 - 1;
    int ww = w0 + tw, hh = h0 + th, dd = d0 + td;
    bool inb = (unsigned)ww < SW && (unsigned)hh < SH && (unsigned)dd < SD;
    int  spc = inb ? (ww * SH + hh) * SD + dd : sp;
    float am = inb ? 1.f : 0.f;
    acc += wc[t] * base[spc] * am;              // branch-free
  }
  float xv = base[sp];
  float s1 = g1[c] * rsqrtf(v1[c] + EPSV);
  float o1 = (acc - m1[c]) * s1 + b1[c];
  float s2 = g2[c] * rsqrtf(v2[c] + EPSV);
  float o2 = (xv * pw[c] - m2[c]) * s2 + b2[c];
  Out[idx] = o1 + o2 + xv;
}

// --------------- LayerNorm over C=64 per token (wave per row) ---------------
__global__ void ln_kernel(const float* __restrict__ Bf, float* __restrict__ S,
                          const float* __restrict__ lg, const float* __restrict__ lb)
{
  int lane = threadIdx.x & 31;
  int wave = threadIdx.x >> 5;
  int row  = blockIdx.x * 8 + wave;             // 0..NROW-1
  int b = row / LSP, sp = row % LSP;
  const float* base = Bf + (long)b * C_OUT * LSP + sp;
  float v0 = base[(long)lane * LSP];
  float v1 = base[(long)(lane + 32) * LSP];
  float s = v0 + v1, s2 = v0 * v0 + v1 * v1;
  for (int o = 16; o; o >>= 1) {
    s  += __shfl_xor(s, o, 32);
    s2 += __shfl_xor(s2, o, 32);
  }
  float mu = s / 64.f;
  float rs = rsqrtf(s2 / 64.f - mu * mu + EPSV);
  float* out = S + (long)row * C_OUT;
  out[lane]      = (v0 - mu) * rs * lg[lane]      + lb[lane];
  out[lane + 32] = (v1 - mu) * rs * lg[lane + 32] + lb[lane + 32];
}

// ---------------- causal depthwise conv1d (D_CONV=4) + SiLU -----------------
__global__ void conv1d_silu(const float* __restrict__ XZ,
                            const float* __restrict__ wv,
                            const float* __restrict__ bv,
                            float* __restrict__ U)
{
  int idx = blockIdx.x * 256 + threadIdx.x;     // (b,t,d)
  int d = idx % DIN;
  int t = (idx / DIN) % LSP;
  int b = idx / (DIN * LSP);
  const float* col = XZ + (long)b * LSP * (2 * DIN) + d;   // xm(b,t,d)
  float acc = bv[d];
#pragma unroll
  for (int j = 0; j < 4; ++j) {
    int  tt = t - 3 + j;
    int  tc = (tt >= 0) ? tt : 0;
    float m = (tt >= 0) ? 1.f : 0.f;
    acc += wv[d * 4 + j] * col[(long)tc * (2 * DIN)] * m;  // branch-free
  }
  float sig = 1.f / (1.f + __expf(-acc));
  U[((long)b * LSP + t) * DIN + d] = acc * sig;
}

// ---------------- selective scan: 16-lane group per (b,d) channel -----------
// NOTE: Y aliases DT (read-before-write same iteration) -> no restrict here.
__global__ void scan_kernel(const float* DT, const float* __restrict__ XD,
                            const float* __restrict__ U, const float* __restrict__ XZ,
                            const float* __restrict__ Alog, const float* __restrict__ Dp,
                            float* Y)
{
  int tid = blockIdx.x * 256 + threadIdx.x;
  int grp = tid >> 4;                           // 0..255 = (b,d)
  int s   = tid & 15;                           // state index
  int b   = grp >> 7;
  int d   = grp & 127;
  float Ac = -__expf(Alog[d * DST + s]);
  float Dv = Dp[d];
  float h  = 0.f;
  const long rowD  = (long)b * LSP * DIN + d;
  const long rowXD = (long)b * LSP * 36;
  const long rowXZ = (long)b * LSP * (2 * DIN) + DIN + d;
  for (int t = 0; t < LSP; ++t) {
    if (t + 8 < LSP) {                          // hide latency on serial chain
      __builtin_prefetch(&DT[rowD + (long)(t + 8) * DIN], 0, 1);
      __builtin_prefetch(&XD[rowXD + (long)(t + 8) * 36 + DTR + s], 0, 1);
    }
    float dtp = DT[rowD + (long)t * DIN];
    float dtv = (dtp > 20.f) ? dtp : log1pf(__expf(dtp));   // softplus
    float uv  = U[rowD + (long)t * DIN];
    float Bv  = XD[rowXD + (long)t * 36 + DTR + s];
    float Cv  = XD[rowXD + (long)t * 36 + DTR + DST + s];
    h = h * __expf(dtv * Ac) + dtv * uv * Bv;
    float p = h * Cv;
    p += __shfl_xor(p, 8, 16);
    p += __shfl_xor(p, 4, 16);
    p += __shfl_xor(p, 2, 16);
    p += __shfl_xor(p, 1, 16);
    if (s == 0) {
      float zz = XZ[rowXZ + (long)t * (2 * DIN)];
      float y  = (p + uv * Dv) * (zz / (1.f + __expf(-zz)));  // y * silu(z)
      Y[rowD + (long)t * DIN] = y;
    }
  }
}

// ---------------- final: out = identity + bn'd cfe2 (layout transpose) ------
__global__ void final_add(const float* __restrict__ Iden,
                          const float* __restrict__ C2,
                          float* __restrict__ Outp)
{
  int idx = blockIdx.x * 256 + threadIdx.x;     // NCDHW
  int sp = idx % LSP;
  int c  = (idx / LSP) % C_OUT;
  int b  = idx / (C_OUT * LSP);
  Outp[idx] = Iden[idx] + C2[((long)b * LSP + sp) * C_OUT + c];
}

// ---------------------------------------------------------------------------
extern "C" void kernel_launch(void* const* d_in, const int* in_sizes, int n_in,
                              void* d_out, int out_size, void* d_ws, size_t ws_size,
                              hipStream_t stream)
{
  const float* x          = (const float*)d_in[0];
  const float* conv_w     = (const float*)d_in[1];
  const float* conv_b     = (const float*)d_in[2];
  const float* gn_g       = (const float*)d_in[3];
  const float* gn_b       = (const float*)d_in[4];
  const float* sfe_conv_w = (const float*)d_in[5];
  const float* sfe_bn1_g  = (const float*)d_in[6];
  const float* sfe_bn1_b  = (const float*)d_in[7];
  const float* sfe_bn1_m  = (const float*)d_in[8];
  const float* sfe_bn1_v  = (const float*)d_in[9];
  const float* sfe_conv1_w= (const float*)d_in[10];
  const float* sfe_bn2_g  = (const float*)d_in[11];
  const float* sfe_bn2_b  = (const float*)d_in[12];
  const float* sfe_bn2_m  = (const float*)d_in[13];
  const float* sfe_bn2_v  = (const float*)d_in[14];
  const float* ln_g       = (const float*)d_in[15];
  const float* ln_b       = (const float*)d_in[16];
  const float* in_proj_w  = (const float*)d_in[17];
  const float* conv1d_w   = (const float*)d_in[18];
  const float* conv1d_b   = (const float*)d_in[19];
  const float* x_proj_w   = (const float*)d_in[20];
  const float* dt_proj_w  = (const float*)d_in[21];
  const float* dt_proj_b  = (const float*)d_in[22];
  const float* A_log      = (const float*)d_in[23];
  const float* Dp         = (const float*)d_in[24];
  const float* out_proj_w = (const float*)d_in[25];
  const float* cfe1_w     = (const float*)d_in[26];
  const float* cfe_bn1_g  = (const float*)d_in[27];
  const float* cfe_bn1_b  = (const float*)d_in[28];
  const float* cfe_bn1_m  = (const float*)d_in[29];
  const float* cfe_bn1_v  = (const float*)d_in[30];
  const float* cfe2_w     = (const float*)d_in[31];
  const float* cfe_bn2_g  = (const float*)d_in[32];
  const float* cfe_bn2_b  = (const float*)d_in[33];
  const float* cfe_bn2_m  = (const float*)d_in[34];
  const float* cfe_bn2_v  = (const float*)d_in[35];

  // Workspace layout (floats), ~101 MB total; Y aliases DT (read-before-write)
  float* ws  = (float*)d_ws;
  float* A   = ws;                                  // 2M   identity (NCDHW)
  float* Bf  = A   + (size_t)NB * C_OUT * LSP;      // 2M   sfe out / cfe2 out
  float* S   = Bf  + (size_t)NB * C_OUT * LSP;      // 2M   LN out / out_proj out
  float* XZ  = S   + (size_t)NROW * C_OUT;          // 8M   in_proj out / cfe1 out
  float* U   = XZ  + (size_t)NROW * 2 * DIN;        // 4M   silu(conv1d)
  float* XD  = U   + (size_t)NROW * DIN;            // 1.2M x_dbl
  float* DTb = XD  + (size_t)NROW * 36;             // 4M   dt pre-softplus
  float* XT  = DTb + (size_t)NROW * DIN;            // 1M   x transposed [row][ci]
  float* W2  = XT  + (size_t)NROW * C_IN;           // 55K  conv_w [tap][ci][co]
  float* st  = W2  + (size_t)27 * C_IN * C_OUT;     // 128  gn stats
  float* gmean = st, *grstd = st + 64;
  float* O1 = S;                                    // reuse (S dead post in_proj)
  float* C1 = XZ;                                   // reuse (XZ dead post scan)
  float* C2 = Bf;                                   // reuse (Bf dead post LN)
  float* Yb = DTb;                                  // alias (read-before-write)
  (void)in_sizes; (void)n_in; (void)out_size; (void)ws_size;

  const int NE = NB * C_OUT * LSP;                  // 2,097,152

  // 0) layout prep for implicit-GEMM conv
  xpose_kernel<<<(NB * C_IN * LSP) / 256, 256, 0, stream>>>(x, XT);
  wpose_kernel<<<(C_OUT * C_IN * 27 + 255) / 256, 256, 0, stream>>>(conv_w, W2);
  // 1) conv3d 3x3x3 + bias (implicit GEMM, WMMA)
  conv3d_wmma<<<NROW / 16, 128, 0, stream>>>(XT, W2, conv_b, A);
  // 2) groupnorm + relu  -> A is 'identity'
  gn_stats<<<64, 256, 0, stream>>>(A, gmean, grstd);
  gn_apply<<<NE / 256, 256, 0, stream>>>(A, gmean, grstd, gn_g, gn_b);
  // 3) SFE residual block
  sfe_kernel<<<NE / 256, 256, 0, stream>>>(A, Bf, sfe_conv_w,
      sfe_bn1_g, sfe_bn1_b, sfe_bn1_m, sfe_bn1_v,
      sfe_conv1_w, sfe_bn2_g, sfe_bn2_b, sfe_bn2_m, sfe_bn2_v);
  // 4) layernorm over channels -> S [NROW,64]
  ln_kernel<<<NROW / 8, 256, 0, stream>>>(Bf, S, ln_g, ln_b);
  // 5) in_proj: [NROW,64] x [256,64]^T -> XZ [NROW,256]
  gemm_f32_wmma<false><<<dim3(NROW / 16, 4), 128, 0, stream>>>(S, C_OUT,
      in_proj_w, XZ, 2 * DIN, 2 * DIN, C_OUT,
      nullptr, nullptr, nullptr, nullptr, nullptr, 0);
  // 6) causal conv1d + silu -> U [NROW,128]
  conv1d_silu<<<(NROW * DIN) / 256, 256, 0, stream>>>(XZ, conv1d_w, conv1d_b, U);
  // 7) x_proj: [NROW,128] x [36,128]^T -> XD [NROW,36]   (N-guarded)
  gemm_f32_wmma<true><<<dim3(NROW / 16, 1), 128, 0, stream>>>(U, DIN,
      x_proj_w, XD, 36, 36, DIN,
      nullptr, nullptr, nullptr, nullptr, nullptr, 0);
  // 8) dt_proj: [NROW,4] x [128,4]^T + b -> DTb [NROW,128]
  gemm_f32_wmma<false><<<dim3(NROW / 16, 2), 128, 0, stream>>>(XD, 36,
      dt_proj_w, DTb, DIN, DIN, DTR,
      dt_proj_b, nullptr, nullptr, nullptr, nullptr, 0);
  // 9) selective scan (+ skip, * silu(z)) -> Yb [NROW,128]
  scan_kernel<<<16, 256, 0, stream>>>(DTb, XD, U, XZ, A_log, Dp, Yb);
  // 10) out_proj: [NROW,128] x [64,128]^T -> O1 [NROW,64]
  gemm_f32_wmma<false><<<dim3(NROW / 16, 1), 128, 0, stream>>>(Yb, DIN,
      out_proj_w, O1, C_OUT, C_OUT, DIN,
      nullptr, nullptr, nullptr, nullptr, nullptr, 0);
  // 11) cfe1 (1x1x1) + BN + relu -> C1 [NROW,256]
  gemm_f32_wmma<false><<<dim3(NROW / 16, 4), 128, 0, stream>>>(O1, C_OUT,
      cfe1_w, C1, 4 * C_OUT, 4 * C_OUT, C_OUT,
      nullptr, cfe_bn1_g, cfe_bn1_b, cfe_bn1_m, cfe_bn1_v, 1);
  // 12) cfe2 (1x1x1) + BN -> C2 [NROW,64]
  gemm_f32_wmma<false><<<dim3(NROW / 16, 1), 128, 0, stream>>>(C1, 4 * C_OUT,
      cfe2_w, C2, C_OUT, C_OUT, 4 * C_OUT,
      nullptr, cfe_bn2_g, cfe_bn2_b, cfe_bn2_m, cfe_bn2_v, 0);
  // 13) out = identity + C2 (transpose back to NCDHW)
  final_add<<<NE / 256, 256, 0, stream>>>(A, C2, (float*)d_out);
}